// GATLayer_48825188220995
// MI455X (gfx1250) — compile-verified
//
#include <hip/hip_runtime.h>
#include <hip/hip_bf16.h>
#include <math.h>

typedef float v2f __attribute__((ext_vector_type(2)));
typedef float v8f __attribute__((ext_vector_type(8)));

#define NEG_SLOPE 0.2f
#define EPS_F 1e-16f

// ds_swizzle_b32 group-of-32 XOR butterfly (offset = xor<<10 | or<<5 | and)
#define SWIZ_XOR1  0x041F
#define SWIZ_XOR2  0x081F
#define SWIZ_XOR4  0x101F
#define SWIZ_XOR8  0x201F
#define SWIZ_XOR16 0x401F
#define SWZ(v, imm) __int_as_float(__builtin_amdgcn_ds_swizzle(__float_as_int(v), imm))

// sum across each 16-lane half (xor masks 1..8 never cross the half boundary)
__device__ __forceinline__ float reduce16_add(float v) {
    v += SWZ(v, SWIZ_XOR1);
    v += SWZ(v, SWIZ_XOR2);
    v += SWZ(v, SWIZ_XOR4);
    v += SWZ(v, SWIZ_XOR8);
    return v;
}
// max across all 32 lanes
__device__ __forceinline__ float reduce32_max(float v) {
    v = fmaxf(v, SWZ(v, SWIZ_XOR1));
    v = fmaxf(v, SWZ(v, SWIZ_XOR2));
    v = fmaxf(v, SWZ(v, SWIZ_XOR4));
    v = fmaxf(v, SWZ(v, SWIZ_XOR8));
    v = fmaxf(v, SWZ(v, SWIZ_XOR16));
    return v;
}

// ---------- order-preserving float <-> uint key for atomic max ----------
__device__ __forceinline__ unsigned int fkey(float f) {
    unsigned int u = __float_as_uint(f);
    return (u & 0x80000000u) ? ~u : (u | 0x80000000u);
}
__device__ __forceinline__ float funkey(unsigned int k) {
    return __uint_as_float((k & 0x80000000u) ? (k & 0x7FFFFFFFu) : ~k);
}

// =====================================================================
// Kernel 1: h = x @ W  (fp32 WMMA 16x16x4), fused s_src/s_trg reduction
//   block = 256 threads = 8 waves; wave w -> head w (cols w*16..w*16+15)
//   blockIdx.x -> 16-row M tile
// =====================================================================
__global__ __launch_bounds__(256) void gat_gemm_wmma(
    const float* __restrict__ x, const float* __restrict__ W,
    const float* __restrict__ a_src, const float* __restrict__ a_trg,
    float* __restrict__ hbuf, float* __restrict__ s_src, float* __restrict__ s_trg,
    int N)
{
    const int wave  = threadIdx.x >> 5;      // 0..7 : head / N-tile
    const int lane  = threadIdx.x & 31;
    const int mBase = blockIdx.x * 16;
    const int nBase = wave * 16;

    const int f    = lane & 15;              // col-within-tile / K-feature lane
    const int half = lane >> 4;              // 0 or 1
    const int ksub = half * 2;               // K offset pair for A/B frags

    // clamp row for loads so EXEC stays all-ones around the WMMA ops
    int rowA = mBase + f;
    if (rowA > N - 1) rowA = N - 1;
    const float* __restrict__ xrow = x + (size_t)rowA * 128;

    v8f c = {};
    #pragma unroll 8
    for (int k = 0; k < 128; k += 4) {
        // A fragment: lane holds x[rowA][k+ksub], x[rowA][k+ksub+1] (contiguous -> b64 load)
        v2f a;
        a.x = xrow[k + ksub];
        a.y = xrow[k + ksub + 1];
        // B fragment: lane holds W[k+ksub][nBase+f], W[k+ksub+1][nBase+f]
        v2f b;
        b.x = W[(size_t)(k + ksub) * 128 + nBase + f];
        b.y = W[(size_t)(k + ksub + 1) * 128 + nBase + f];
        c = __builtin_amdgcn_wmma_f32_16x16x4_f32(
                /*neg_a=*/false, a, /*neg_b=*/false, b,
                /*c_mod=*/(short)0, c, /*reuse_a=*/false, /*reuse_b=*/false);
    }

    // attention vectors for this head; f = feature index within head
    const float asv = a_src[nBase + f];
    const float atv = a_trg[nBase + f];

    #pragma unroll
    for (int r = 0; r < 8; ++r) {
        const int m = mBase + r + half * 8;      // C/D layout: VGPR r -> row r (+8 for hi half)
        const float hv = c[r];
        if (m < N) hbuf[(size_t)m * 128 + nBase + f] = hv;

        // 4-step swizzle butterfly reduce over the 16 feature lanes
        const float ps = reduce16_add(hv * asv);
        const float pt = reduce16_add(hv * atv);
        if (f == 0 && m < N) {
            s_src[(size_t)m * 8 + wave] = ps;
            s_trg[(size_t)m * 8 + wave] = pt;
        }
    }
}

// =====================================================================
// Kernel 2: per-edge leaky-relu scores + global max (ordered-uint atomic)
// =====================================================================
__global__ __launch_bounds__(256) void gat_edge_scores(
    const int* __restrict__ ei, int E,
    const float* __restrict__ s_src, const float* __restrict__ s_trg,
    float* __restrict__ ebuf, unsigned int* __restrict__ gmax)
{
    __shared__ unsigned int smax;
    if (threadIdx.x == 0) smax = 0u;
    __syncthreads();

    const int e = blockIdx.x * blockDim.x + threadIdx.x;
    float lmax = -INFINITY;
    if (e < E) {
        const int src = ei[e];
        const int trg = ei[E + e];
        const float* __restrict__ ss = s_src + (size_t)src * 8;
        const float* __restrict__ st = s_trg + (size_t)trg * 8;
        #pragma unroll
        for (int hh = 0; hh < 8; ++hh) {
            float v = ss[hh] + st[hh];
            v = (v >= 0.f) ? v : NEG_SLOPE * v;
            ebuf[(size_t)e * 8 + hh] = v;
            lmax = fmaxf(lmax, v);
        }
    }
    lmax = reduce32_max(lmax);
    if ((threadIdx.x & 31) == 0) atomicMax(&smax, fkey(lmax));
    __syncthreads();
    if (threadIdx.x == 0) atomicMax(gmax, smax);
}

// =====================================================================
// Kernel 3: e = exp(score - gmax); denom[trg,h] += e
// =====================================================================
__global__ __launch_bounds__(256) void gat_edge_exp(
    const int* __restrict__ ei, int E,
    float* __restrict__ ebuf, float* __restrict__ denom,
    const unsigned int* __restrict__ gmax)
{
    const int e = blockIdx.x * blockDim.x + threadIdx.x;
    if (e >= E) return;
    const float mx = funkey(*gmax);
    const int trg = ei[E + e];
    #pragma unroll
    for (int hh = 0; hh < 8; ++hh) {
        const float v = expf(ebuf[(size_t)e * 8 + hh] - mx);
        ebuf[(size_t)e * 8 + hh] = v;
        atomicAdd(&denom[(size_t)trg * 8 + hh], v);
    }
}

// =====================================================================
// Kernel 4: out[trg,h,:] += alpha * h[src,h,:]  (thread per edge-head)
// =====================================================================
__global__ __launch_bounds__(256) void gat_aggregate(
    const int* __restrict__ ei, int E,
    const float* __restrict__ hbuf, const float* __restrict__ ebuf,
    const float* __restrict__ denom, float* __restrict__ out)
{
    const int t = blockIdx.x * blockDim.x + threadIdx.x;   // edge*8 + head
    if (t >= E * 8) return;
    const int e  = t >> 3;
    const int hh = t & 7;
    const int src = ei[e];
    const int trg = ei[E + e];

    const float alpha = ebuf[t] / (denom[(size_t)trg * 8 + hh] + EPS_F);
    const float4* __restrict__ hp = (const float4*)(hbuf + (size_t)src * 128 + hh * 16);
    float* __restrict__ op = out + (size_t)trg * 128 + hh * 16;
    #pragma unroll
    for (int q = 0; q < 4; ++q) {
        const float4 v = hp[q];
        atomicAdd(op + q * 4 + 0, v.x * alpha);
        atomicAdd(op + q * 4 + 1, v.y * alpha);
        atomicAdd(op + q * 4 + 2, v.z * alpha);
        atomicAdd(op + q * 4 + 3, v.w * alpha);
    }
}

// =====================================================================
// Kernel 5: out = elu(out + h + bias)
// =====================================================================
__global__ __launch_bounds__(256) void gat_epilogue(
    const float* __restrict__ hbuf, const float* __restrict__ bias,
    float* __restrict__ out, int total)
{
    const int i = blockIdx.x * blockDim.x + threadIdx.x;
    if (i >= total) return;
    float v = out[i] + hbuf[i] + bias[i & 127];
    out[i] = (v > 0.f) ? v : expm1f(v);
}

// =====================================================================
extern "C" void kernel_launch(void* const* d_in, const int* in_sizes, int n_in,
                              void* d_out, int out_size, void* d_ws, size_t ws_size,
                              hipStream_t stream)
{
    const float* x      = (const float*)d_in[0];
    const int*   ei     = (const int*)  d_in[1];
    const float* W      = (const float*)d_in[2];
    const float* a_src  = (const float*)d_in[3];
    const float* a_trg  = (const float*)d_in[4];
    const float* bias   = (const float*)d_in[5];

    const int N = in_sizes[0] / 128;
    const int E = in_sizes[1] / 2;

    // workspace layout (floats)
    float* hbuf  = (float*)d_ws;                      // N*128
    float* ssrc  = hbuf  + (size_t)N * 128;           // N*8
    float* strg  = ssrc  + (size_t)N * 8;             // N*8
    float* ebuf  = strg  + (size_t)N * 8;             // E*8
    float* denom = ebuf  + (size_t)E * 8;             // N*8
    unsigned int* gmax = (unsigned int*)(denom + (size_t)N * 8); // 1

    // zero denom + gmax (0 == identity for ordered-key max) and output accumulator
    hipMemsetAsync(denom, 0, (size_t)N * 8 * sizeof(float) + sizeof(unsigned int), stream);
    hipMemsetAsync(d_out, 0, (size_t)out_size * sizeof(float), stream);

    const int mtiles = (N + 15) / 16;
    gat_gemm_wmma<<<mtiles, 256, 0, stream>>>(x, W, a_src, a_trg, hbuf, ssrc, strg, N);

    gat_edge_scores<<<(E + 255) / 256, 256, 0, stream>>>(ei, E, ssrc, strg, ebuf, gmax);

    gat_edge_exp<<<(E + 255) / 256, 256, 0, stream>>>(ei, E, ebuf, denom, gmax);

    gat_aggregate<<<(E * 8 + 255) / 256, 256, 0, stream>>>(ei, E, hbuf, ebuf, denom, (float*)d_out);

    const int total = N * 128;
    gat_epilogue<<<(total + 255) / 256, 256, 0, stream>>>(hbuf, bias, (float*)d_out, total);
}